// BIMPM_36661840839451
// MI455X (gfx1250) — compile-verified
//
#include <hip/hip_runtime.h>

// ---- problem constants ----
#define B_    32
#define S_    60
#define H_    100
#define L_    20
#define D_    300
#define G_    400   // 4*H
#define T_    60
#define CLS_  3
#define AGGIN_ 160  // 8*L

typedef __attribute__((ext_vector_type(2))) float v2f;
typedef __attribute__((ext_vector_type(8))) float v8f;

__device__ __forceinline__ float sigmoidf_(float x) { return 1.f / (1.f + __expf(-x)); }

__device__ __forceinline__ v2f ld2g(const float* p) {  // 8B-aligned global load
  return *(const v2f*)p;
}
__device__ __forceinline__ v2f ld2s(const float* p) {  // 8B-aligned LDS load
  return *(const v2f*)p;
}

// ---------------------------------------------------------------------------
// Embedding gather: out[row, :] = emb[idx[row], :]
// ---------------------------------------------------------------------------
__global__ void gather_emb_kernel(const int* __restrict__ idx,
                                  const float* __restrict__ emb,
                                  float* __restrict__ out) {
  int row = blockIdx.x;
  int w = idx[row];
  for (int d = threadIdx.x; d < D_; d += blockDim.x)
    out[(size_t)row * D_ + d] = emb[(size_t)w * D_ + d];
}

// ---------------------------------------------------------------------------
// Generic fp32 WMMA GEMM:  C[M,N] = A[M,K] @ W[N,K]^T + bias[N]
// One wave per 16x16 tile; K stepped by 8 with two independent accumulators
// (v_wmma_f32_16x16x4_f32). Branch-free inner loop: OOB rows are clamped --
// garbage lands only in accumulator slots that the guarded epilogue never
// stores. Requires K % 4 == 0 (true for 300/160/100/60).
// ---------------------------------------------------------------------------
__global__ void gemm_bias_wmma_kernel(const float* __restrict__ A,
                                      const float* __restrict__ W,
                                      const float* __restrict__ bias,
                                      float* __restrict__ C,
                                      int M, int N, int K) {
  int lane = threadIdx.x;              // blockDim.x == 32 (one wave)
  int nt = blockIdx.x, mt = blockIdx.y;
  int m = mt * 16 + (lane & 15);
  int n = nt * 16 + (lane & 15);
  int khalf = (lane >> 4) * 2;         // 0 or 2
  int mc = (m < M) ? m : (M - 1);      // clamp: safe address, result masked by store guard
  int nc = (n < N) ? n : (N - 1);
  const float* Ar = A + (size_t)mc * K + khalf;
  const float* Wr = W + (size_t)nc * K + khalf;
  __builtin_prefetch(Wr, 0, 1);
  __builtin_prefetch(Ar, 0, 1);
  v8f acc0 = {}, acc1 = {};
  int k = 0;
  for (; k + 8 <= K; k += 8) {
    v2f a0 = ld2g(Ar + k),     b0 = ld2g(Wr + k);
    v2f a1 = ld2g(Ar + k + 4), b1 = ld2g(Wr + k + 4);
    acc0 = __builtin_amdgcn_wmma_f32_16x16x4_f32(false, a0, false, b0,
                                                 (short)0, acc0, false, false);
    acc1 = __builtin_amdgcn_wmma_f32_16x16x4_f32(false, a1, false, b1,
                                                 (short)0, acc1, false, false);
  }
  if (k < K) {   // K % 8 == 4 tail
    v2f a0 = ld2g(Ar + k), b0 = ld2g(Wr + k);
    acc0 = __builtin_amdgcn_wmma_f32_16x16x4_f32(false, a0, false, b0,
                                                 (short)0, acc0, false, false);
  }
  acc0 = acc0 + acc1;
  int rowbase = mt * 16 + ((lane >> 4) << 3);
  int col = nt * 16 + (lane & 15);
  if (col < N) {
    float bv = bias ? bias[col] : 0.f;
    if (rowbase + 7 < M) {
#pragma unroll
      for (int r = 0; r < 8; ++r)
        C[(size_t)(rowbase + r) * N + col] = acc0[r] + bv;
    } else {
#pragma unroll
      for (int r = 0; r < 8; ++r)
        if (rowbase + r < M) C[(size_t)(rowbase + r) * N + col] = acc0[r] + bv;
    }
  }
}

// ---------------------------------------------------------------------------
// Persistent LSTM (one workgroup = one direction over all 60 steps).
// xp (B,T,G) already contains x@Wih^T + bih.  h kept in LDS, c in registers.
// Per step: g = h @ Whh^T (WMMA, 50 exact tiles / 8 waves) + xp[:,tt,:] + bhh,
// then gates.  reverse=1 walks time backward (hs written at original index).
// LDS: 3200*4 + 12800*4 = 64000 bytes.
// ---------------------------------------------------------------------------
__global__ void __launch_bounds__(256)
lstm_wmma_kernel(const float* __restrict__ xp,
                 const float* __restrict__ Whh,   // (G,H)
                 const float* __restrict__ bhh,   // (G)
                 float* __restrict__ hs_out,      // (B,T,H) or null
                 float* __restrict__ hT_out,      // (B,H) or null
                 int reverse) {
  __shared__ float h_s[B_ * H_];
  __shared__ float g_s[B_ * G_];
  int tid = threadIdx.x;
  int lane = tid & 31;
  int wave = tid >> 5;                 // 0..7
  float c_reg[13];
#pragma unroll
  for (int e = 0; e < 13; ++e) c_reg[e] = 0.f;
  for (int e = tid; e < B_ * H_; e += 256) h_s[e] = 0.f;
  __syncthreads();

  for (int t = 0; t < T_; ++t) {
    int tt = reverse ? (T_ - 1 - t) : t;
    // ---- g = h_s(32x100) @ Whh^T(100x400); all tiles exact ----
    for (int tile = wave; tile < 50; tile += 8) {
      int mt = tile / 25, nt = tile % 25;
      int m = mt * 16 + (lane & 15);
      int n = nt * 16 + (lane & 15);
      int khalf = (lane >> 4) * 2;
      const float* Ar = h_s + m * H_ + khalf;            // LDS
      const float* Wr = Whh + (size_t)n * H_ + khalf;    // global (L2-resident)
      __builtin_prefetch(Wr, 0, 1);
      v8f acc0 = {}, acc1 = {};
      int k = 0;
      for (; k + 8 <= H_; k += 8) {
        v2f a0 = ld2s(Ar + k),     b0 = ld2g(Wr + k);
        v2f a1 = ld2s(Ar + k + 4), b1 = ld2g(Wr + k + 4);
        acc0 = __builtin_amdgcn_wmma_f32_16x16x4_f32(false, a0, false, b0,
                                                     (short)0, acc0, false, false);
        acc1 = __builtin_amdgcn_wmma_f32_16x16x4_f32(false, a1, false, b1,
                                                     (short)0, acc1, false, false);
      }
      {  // H_ % 8 == 4 tail
        v2f a0 = ld2s(Ar + k), b0 = ld2g(Wr + k);
        acc0 = __builtin_amdgcn_wmma_f32_16x16x4_f32(false, a0, false, b0,
                                                     (short)0, acc0, false, false);
      }
      acc0 = acc0 + acc1;
      int rowbase = mt * 16 + ((lane >> 4) << 3);
      int col = nt * 16 + (lane & 15);
#pragma unroll
      for (int r = 0; r < 8; ++r) {
        int row = rowbase + r;   // batch index, always < 32
        g_s[row * G_ + col] =
            acc0[r] + xp[((size_t)row * T_ + tt) * G_ + col] + bhh[col];
      }
    }
    __syncthreads();
    // ---- gates (i,f,g,o blocks of 100) ----
    {
      int idx = tid;
#pragma unroll
      for (int e = 0; e < 13; ++e, idx += 256) {
        if (idx < B_ * H_) {
          int b = idx / H_, hh = idx - b * H_;
          float gi = g_s[b * G_ + hh];
          float gf = g_s[b * G_ + H_ + hh];
          float gg = g_s[b * G_ + 2 * H_ + hh];
          float go = g_s[b * G_ + 3 * H_ + hh];
          float c = sigmoidf_(gf) * c_reg[e] + sigmoidf_(gi) * tanhf(gg);
          c_reg[e] = c;
          float hv = sigmoidf_(go) * tanhf(c);
          h_s[idx] = hv;
          if (hs_out) hs_out[((size_t)b * T_ + tt) * H_ + hh] = hv;
        }
      }
    }
    __syncthreads();
  }
  if (hT_out)
    for (int e = tid; e < B_ * H_; e += 256) hT_out[e] = h_s[e];
}

// ---------------------------------------------------------------------------
// mp_single: cosine(w_l * v1[b,i,:], w_l * u[b,:]) -> mv[b, i, blk*L + l]
// u[b,h] = U[b*u_bstride + u_off + h]
// ---------------------------------------------------------------------------
__global__ void mp_single_kernel(const float* __restrict__ v1,
                                 const float* __restrict__ U,
                                 int u_bstride, int u_off,
                                 const float* __restrict__ w,
                                 float* __restrict__ mv, int blk) {
  int i = blockIdx.x, b = blockIdx.y, l = threadIdx.x;
  if (l >= L_) return;
  const float* v = v1 + ((size_t)b * S_ + i) * H_;
  const float* u = U + (size_t)b * u_bstride + u_off;
  float num = 0.f, nv = 0.f, nu = 0.f;
  for (int hh = 0; hh < H_; ++hh) {
    float w2 = w[hh * L_ + l]; w2 *= w2;
    float a = v[hh], c = u[hh];
    num += a * w2 * c; nv += a * a * w2; nu += c * c * w2;
  }
  mv[((size_t)b * S_ + i) * (8 * L_) + blk * L_ + l] = num / (sqrtf(nv) * sqrtf(nu));
}

// per-perspective norms for pairwise matching: nbuf[b,i,l] = ||w_l * v[b,i,:]||
__global__ void mp_norm_kernel(const float* __restrict__ v,
                               const float* __restrict__ w,
                               float* __restrict__ nbuf) {
  int i = blockIdx.x, b = blockIdx.y, l = threadIdx.x;
  if (l >= L_) return;
  const float* vv = v + ((size_t)b * S_ + i) * H_;
  float nv = 0.f;
  for (int hh = 0; hh < H_; ++hh) {
    float w2 = w[hh * L_ + l]; w2 *= w2;
    nv += vv[hh] * vv[hh] * w2;
  }
  nbuf[((size_t)b * S_ + i) * L_ + l] = sqrtf(nv);
}

// ---------------------------------------------------------------------------
// Pairwise numerator: num[b,l,i,j] = sum_h v1[b,i,h]*w[h,l]^2*v2[b,j,h]
// 640 batched 60x100x60 GEMMs; one wave per 16x16 tile (4x4 tiles of 64).
// Clamped (branch-free) loads; OOB rows/cols masked by the store guard.
// ---------------------------------------------------------------------------
__global__ void mp_pair_wmma_kernel(const float* __restrict__ v1,
                                    const float* __restrict__ v2,
                                    const float* __restrict__ w,
                                    float* __restrict__ num) {
  int lane = threadIdx.x;
  int tile = blockIdx.x;               // 0..15
  int l = blockIdx.y, b = blockIdx.z;
  int mt = tile >> 2, nt = tile & 3;
  int m = mt * 16 + (lane & 15);
  int n = nt * 16 + (lane & 15);
  int khalf = (lane >> 4) * 2;
  int mc = (m < S_) ? m : (S_ - 1);
  int nc = (n < S_) ? n : (S_ - 1);
  const float* V1r = v1 + ((size_t)b * S_ + mc) * H_ + khalf;
  const float* V2r = v2 + ((size_t)b * S_ + nc) * H_ + khalf;
  const float* wl = w + l;             // w[(k)*L + l]
  v8f acc0 = {}, acc1 = {};
  int k = 0;
  for (; k + 8 <= H_; k += 8) {
    int ka = k + khalf;
    float w00 = wl[ka * L_];           w00 *= w00;
    float w01 = wl[(ka + 1) * L_];     w01 *= w01;
    float w10 = wl[(ka + 4) * L_];     w10 *= w10;
    float w11 = wl[(ka + 5) * L_];     w11 *= w11;
    v2f a0 = ld2g(V1r + k);     a0.x *= w00; a0.y *= w01;
    v2f a1 = ld2g(V1r + k + 4); a1.x *= w10; a1.y *= w11;
    v2f b0 = ld2g(V2r + k);
    v2f b1 = ld2g(V2r + k + 4);
    acc0 = __builtin_amdgcn_wmma_f32_16x16x4_f32(false, a0, false, b0,
                                                 (short)0, acc0, false, false);
    acc1 = __builtin_amdgcn_wmma_f32_16x16x4_f32(false, a1, false, b1,
                                                 (short)0, acc1, false, false);
  }
  {  // H_ % 8 == 4 tail
    int ka = k + khalf;
    float w00 = wl[ka * L_];       w00 *= w00;
    float w01 = wl[(ka + 1) * L_]; w01 *= w01;
    v2f a0 = ld2g(V1r + k); a0.x *= w00; a0.y *= w01;
    v2f b0 = ld2g(V2r + k);
    acc0 = __builtin_amdgcn_wmma_f32_16x16x4_f32(false, a0, false, b0,
                                                 (short)0, acc0, false, false);
  }
  acc0 = acc0 + acc1;
  int rowbase = mt * 16 + ((lane >> 4) << 3);
  int col = nt * 16 + (lane & 15);
  float* out = num + ((size_t)b * L_ + l) * S_ * S_;
  if (col < S_) {
#pragma unroll
    for (int r = 0; r < 8; ++r) {
      int row = rowbase + r;
      if (row < S_) out[(size_t)row * S_ + col] = acc0[r];
    }
  }
}

// mv_p_max[b,i,l] = max_j num/(n1[b,i,l]*n2[b,j,l])
__global__ void pair_max_j_kernel(const float* __restrict__ num,
                                  const float* __restrict__ n1,
                                  const float* __restrict__ n2,
                                  float* __restrict__ mvp, int blk) {
  int i = blockIdx.x, b = blockIdx.y, l = threadIdx.x;
  if (l >= L_) return;
  const float* row = num + (((size_t)b * L_ + l) * S_ + i) * S_;
  float n1v = n1[((size_t)b * S_ + i) * L_ + l];
  float mx = -3.4e38f;
  for (int j = 0; j < S_; ++j)
    mx = fmaxf(mx, row[j] / (n1v * n2[((size_t)b * S_ + j) * L_ + l]));
  mvp[((size_t)b * S_ + i) * (8 * L_) + blk * L_ + l] = mx;
}

// mv_h_max[b,j,l] = max_i num/(n1[b,i,l]*n2[b,j,l])
__global__ void pair_max_i_kernel(const float* __restrict__ num,
                                  const float* __restrict__ n1,
                                  const float* __restrict__ n2,
                                  float* __restrict__ mvh, int blk) {
  int j = blockIdx.x, b = blockIdx.y, l = threadIdx.x;
  if (l >= L_) return;
  const float* col = num + ((size_t)b * L_ + l) * S_ * S_ + j;
  float n2v = n2[((size_t)b * S_ + j) * L_ + l];
  float mx = -3.4e38f;
  for (int i = 0; i < S_; ++i)
    mx = fmaxf(mx, col[(size_t)i * S_] / (n1[((size_t)b * S_ + i) * L_ + l] * n2v));
  mvh[((size_t)b * S_ + j) * (8 * L_) + blk * L_ + l] = mx;
}

// ---------------------------------------------------------------------------
// Attention: only row 0 / col 0 of the cosine matrix are consumed downstream.
// t==0: arow[b,j] = cos(cp[b,0], ch[b,j]); t==1: acol[b,j] = cos(cp[b,j], ch[b,0])
// ---------------------------------------------------------------------------
__global__ void attn_rowcol_kernel(const float* __restrict__ cp,
                                   const float* __restrict__ ch,
                                   float* __restrict__ arow,
                                   float* __restrict__ acol) {
  int j = blockIdx.x, b = blockIdx.y, t = threadIdx.x;
  if (t >= 2) return;
  const float* x0 = (t == 0) ? cp + (size_t)b * S_ * H_ : ch + (size_t)b * S_ * H_;
  const float* y  = (t == 0) ? ch + ((size_t)b * S_ + j) * H_
                             : cp + ((size_t)b * S_ + j) * H_;
  float d = 0.f, nx = 0.f, ny = 0.f;
  for (int hh = 0; hh < H_; ++hh) {
    float a = x0[hh], c = y[hh];
    d += a * c; nx += a * a; ny += c * c;
  }
  float v = d / (sqrtf(nx) * sqrtf(ny));
  if (t == 0) arow[b * S_ + j] = v; else acol[b * S_ + j] = v;
}

__global__ void attn_vecs_kernel(const float* __restrict__ cp,
                                 const float* __restrict__ ch,
                                 const float* __restrict__ arow,
                                 const float* __restrict__ acol,
                                 float* __restrict__ mean_h, float* __restrict__ mean_p,
                                 float* __restrict__ max_h, float* __restrict__ max_p) {
  int hh = blockIdx.x * blockDim.x + threadIdx.x;
  int b = blockIdx.y;
  if (hh >= H_) return;
  float num_h = 0.f, num_p = 0.f, sr = 0.f, sc = 0.f;
  float mxh = -3.4e38f, mxp = -3.4e38f;
  for (int s = 0; s < S_; ++s) {
    float ar = arow[b * S_ + s], ac = acol[b * S_ + s];
    float chv = ch[((size_t)b * S_ + s) * H_ + hh];
    float cpv = cp[((size_t)b * S_ + s) * H_ + hh];
    num_h += ar * chv; sr += ar;
    num_p += ac * cpv; sc += ac;
    mxh = fmaxf(mxh, chv * ar);
    mxp = fmaxf(mxp, cpv * ac);
  }
  mean_h[b * H_ + hh] = num_h / (sr + 1e-10f);
  mean_p[b * H_ + hh] = num_p / (sc + 1e-10f);
  max_h[b * H_ + hh] = mxh;
  max_p[b * H_ + hh] = mxp;
}

// ---------------------------------------------------------------------------
// FC head
// ---------------------------------------------------------------------------
__global__ void fc1_kernel(const float* __restrict__ h0, const float* __restrict__ h1,
                           const float* __restrict__ h2, const float* __restrict__ h3,
                           const float* __restrict__ W, const float* __restrict__ bias,
                           float* __restrict__ out) {
  int o = blockIdx.x * blockDim.x + threadIdx.x;
  int b = blockIdx.y;
  if (o >= 2 * H_) return;
  const float* w = W + (size_t)o * (4 * H_);
  float s = bias[o];
  for (int k = 0; k < H_; ++k) s += h0[b * H_ + k] * w[k];
  for (int k = 0; k < H_; ++k) s += h1[b * H_ + k] * w[H_ + k];
  for (int k = 0; k < H_; ++k) s += h2[b * H_ + k] * w[2 * H_ + k];
  for (int k = 0; k < H_; ++k) s += h3[b * H_ + k] * w[3 * H_ + k];
  out[b * (2 * H_) + o] = tanhf(s);
}

__global__ void fc2_kernel(const float* __restrict__ x, const float* __restrict__ W,
                           const float* __restrict__ bias, float* __restrict__ out) {
  int o = threadIdx.x, b = blockIdx.x;
  if (o >= CLS_) return;
  float s = bias[o];
  for (int k = 0; k < 2 * H_; ++k) s += x[b * (2 * H_) + k] * W[o * (2 * H_) + k];
  out[b * CLS_ + o] = s;
}

// ---------------------------------------------------------------------------
// host launch
// ---------------------------------------------------------------------------
extern "C" void kernel_launch(void* const* d_in, const int* in_sizes, int n_in,
                              void* d_out, int out_size, void* d_ws, size_t ws_size,
                              hipStream_t stream) {
  (void)in_sizes; (void)n_in; (void)out_size; (void)ws_size;
  const int*   p_idx    = (const int*)d_in[0];
  const int*   h_idx    = (const int*)d_in[1];
  const float* word_emb = (const float*)d_in[2];
  const float* cWih_f = (const float*)d_in[3],  *cWhh_f = (const float*)d_in[4];
  const float* cbih_f = (const float*)d_in[5],  *cbhh_f = (const float*)d_in[6];
  const float* cWih_b = (const float*)d_in[7],  *cWhh_b = (const float*)d_in[8];
  const float* cbih_b = (const float*)d_in[9],  *cbhh_b = (const float*)d_in[10];
  const float* aWih_f = (const float*)d_in[11], *aWhh_f = (const float*)d_in[12];
  const float* abih_f = (const float*)d_in[13], *abhh_f = (const float*)d_in[14];
  const float* aWih_b = (const float*)d_in[15], *aWhh_b = (const float*)d_in[16];
  const float* abih_b = (const float*)d_in[17], *abhh_b = (const float*)d_in[18];
  const float* w[9];
  for (int i = 1; i <= 8; ++i) w[i] = (const float*)d_in[18 + i];  // w1..w8
  const float* fc1_W = (const float*)d_in[27], *fc1_b = (const float*)d_in[28];
  const float* fc2_W = (const float*)d_in[29], *fc2_b = (const float*)d_in[30];

  // ---- workspace carve (floats) ----
  float* wsf = (float*)d_ws;
  size_t off = 0;
  auto alloc = [&](size_t n) { float* p = wsf + off; off += n; return p; };
  float* pe = alloc((size_t)B_ * S_ * D_);
  float* he = alloc((size_t)B_ * S_ * D_);
  float* xp_pf = alloc((size_t)B_ * T_ * G_);
  float* xp_pb = alloc((size_t)B_ * T_ * G_);
  float* xp_hf = alloc((size_t)B_ * T_ * G_);
  float* xp_hb = alloc((size_t)B_ * T_ * G_);
  float* cpf = alloc((size_t)B_ * T_ * H_);
  float* cpb = alloc((size_t)B_ * T_ * H_);
  float* chf = alloc((size_t)B_ * T_ * H_);
  float* chb = alloc((size_t)B_ * T_ * H_);
  float* mvp = alloc((size_t)B_ * S_ * 8 * L_);
  float* mvh = alloc((size_t)B_ * S_ * 8 * L_);
  float* numb = alloc((size_t)B_ * L_ * S_ * S_);
  float* n1b = alloc((size_t)B_ * S_ * L_);
  float* n2b = alloc((size_t)B_ * S_ * L_);
  float* arow = alloc((size_t)B_ * S_);
  float* acol = alloc((size_t)B_ * S_);
  float* meanh = alloc((size_t)B_ * H_);
  float* meanp = alloc((size_t)B_ * H_);
  float* maxh = alloc((size_t)B_ * H_);
  float* maxp = alloc((size_t)B_ * H_);
  float* hT_pf = alloc((size_t)B_ * H_);
  float* hT_pb = alloc((size_t)B_ * H_);
  float* hT_hf = alloc((size_t)B_ * H_);
  float* hT_hb = alloc((size_t)B_ * H_);
  float* fc1o = alloc((size_t)B_ * 2 * H_);
  // agg projections reuse the (now dead) ctx projection buffers
  float* xa_pf = xp_pf; float* xa_pb = xp_pb;
  float* xa_hf = xp_hf; float* xa_hb = xp_hb;

  const int MBS = B_ * S_;          // 1920
  dim3 gGemm((G_ + 15) / 16, (MBS + 15) / 16);

  // 1) embedding gather
  gather_emb_kernel<<<MBS, 128, 0, stream>>>(p_idx, word_emb, pe);
  gather_emb_kernel<<<MBS, 128, 0, stream>>>(h_idx, word_emb, he);

  // 2) ctx input projections (WMMA f32)
  gemm_bias_wmma_kernel<<<gGemm, 32, 0, stream>>>(pe, cWih_f, cbih_f, xp_pf, MBS, G_, D_);
  gemm_bias_wmma_kernel<<<gGemm, 32, 0, stream>>>(pe, cWih_b, cbih_b, xp_pb, MBS, G_, D_);
  gemm_bias_wmma_kernel<<<gGemm, 32, 0, stream>>>(he, cWih_f, cbih_f, xp_hf, MBS, G_, D_);
  gemm_bias_wmma_kernel<<<gGemm, 32, 0, stream>>>(he, cWih_b, cbih_b, xp_hb, MBS, G_, D_);

  // 3) ctx BiLSTMs (persistent LDS-state workgroups, WMMA recurrence)
  lstm_wmma_kernel<<<1, 256, 0, stream>>>(xp_pf, cWhh_f, cbhh_f, cpf, nullptr, 0);
  lstm_wmma_kernel<<<1, 256, 0, stream>>>(xp_pb, cWhh_b, cbhh_b, cpb, nullptr, 1);
  lstm_wmma_kernel<<<1, 256, 0, stream>>>(xp_hf, cWhh_f, cbhh_f, chf, nullptr, 0);
  lstm_wmma_kernel<<<1, 256, 0, stream>>>(xp_hb, cWhh_b, cbhh_b, chb, nullptr, 1);

  // 4) matching, forward dir (blocks 0..3, weights w1,w3,w5,w7, last_idx=-1)
  dim3 gSB(S_, B_);
  mp_single_kernel<<<gSB, 32, 0, stream>>>(cpf, chf, S_ * H_, (S_ - 1) * H_, w[1], mvp, 0);
  mp_single_kernel<<<gSB, 32, 0, stream>>>(chf, cpf, S_ * H_, (S_ - 1) * H_, w[1], mvh, 0);
  mp_norm_kernel<<<gSB, 32, 0, stream>>>(cpf, w[3], n1b);
  mp_norm_kernel<<<gSB, 32, 0, stream>>>(chf, w[3], n2b);
  mp_pair_wmma_kernel<<<dim3(16, L_, B_), 32, 0, stream>>>(cpf, chf, w[3], numb);
  pair_max_j_kernel<<<gSB, 32, 0, stream>>>(numb, n1b, n2b, mvp, 1);
  pair_max_i_kernel<<<gSB, 32, 0, stream>>>(numb, n1b, n2b, mvh, 1);
  attn_rowcol_kernel<<<gSB, 2, 0, stream>>>(cpf, chf, arow, acol);
  attn_vecs_kernel<<<dim3((H_ + 63) / 64, B_), 64, 0, stream>>>(cpf, chf, arow, acol,
                                                                meanh, meanp, maxh, maxp);
  mp_single_kernel<<<gSB, 32, 0, stream>>>(cpf, meanh, H_, 0, w[5], mvp, 2);
  mp_single_kernel<<<gSB, 32, 0, stream>>>(chf, meanp, H_, 0, w[5], mvh, 2);
  mp_single_kernel<<<gSB, 32, 0, stream>>>(cpf, maxh, H_, 0, w[7], mvp, 3);
  mp_single_kernel<<<gSB, 32, 0, stream>>>(chf, maxp, H_, 0, w[7], mvh, 3);

  // 5) matching, backward dir (blocks 4..7, weights w2,w4,w6,w8, last_idx=0)
  mp_single_kernel<<<gSB, 32, 0, stream>>>(cpb, chb, S_ * H_, 0, w[2], mvp, 4);
  mp_single_kernel<<<gSB, 32, 0, stream>>>(chb, cpb, S_ * H_, 0, w[2], mvh, 4);
  mp_norm_kernel<<<gSB, 32, 0, stream>>>(cpb, w[4], n1b);
  mp_norm_kernel<<<gSB, 32, 0, stream>>>(chb, w[4], n2b);
  mp_pair_wmma_kernel<<<dim3(16, L_, B_), 32, 0, stream>>>(cpb, chb, w[4], numb);
  pair_max_j_kernel<<<gSB, 32, 0, stream>>>(numb, n1b, n2b, mvp, 5);
  pair_max_i_kernel<<<gSB, 32, 0, stream>>>(numb, n1b, n2b, mvh, 5);
  attn_rowcol_kernel<<<gSB, 2, 0, stream>>>(cpb, chb, arow, acol);
  attn_vecs_kernel<<<dim3((H_ + 63) / 64, B_), 64, 0, stream>>>(cpb, chb, arow, acol,
                                                                meanh, meanp, maxh, maxp);
  mp_single_kernel<<<gSB, 32, 0, stream>>>(cpb, meanh, H_, 0, w[6], mvp, 6);
  mp_single_kernel<<<gSB, 32, 0, stream>>>(chb, meanp, H_, 0, w[6], mvh, 6);
  mp_single_kernel<<<gSB, 32, 0, stream>>>(cpb, maxh, H_, 0, w[8], mvp, 7);
  mp_single_kernel<<<gSB, 32, 0, stream>>>(chb, maxp, H_, 0, w[8], mvh, 7);

  // 6) aggregation input projections (WMMA, K=160)
  gemm_bias_wmma_kernel<<<gGemm, 32, 0, stream>>>(mvp, aWih_f, abih_f, xa_pf, MBS, G_, AGGIN_);
  gemm_bias_wmma_kernel<<<gGemm, 32, 0, stream>>>(mvp, aWih_b, abih_b, xa_pb, MBS, G_, AGGIN_);
  gemm_bias_wmma_kernel<<<gGemm, 32, 0, stream>>>(mvh, aWih_f, abih_f, xa_hf, MBS, G_, AGGIN_);
  gemm_bias_wmma_kernel<<<gGemm, 32, 0, stream>>>(mvh, aWih_b, abih_b, xa_hb, MBS, G_, AGGIN_);

  // 7) aggregation BiLSTMs (final hidden states only)
  lstm_wmma_kernel<<<1, 256, 0, stream>>>(xa_pf, aWhh_f, abhh_f, nullptr, hT_pf, 0);
  lstm_wmma_kernel<<<1, 256, 0, stream>>>(xa_pb, aWhh_b, abhh_b, nullptr, hT_pb, 1);
  lstm_wmma_kernel<<<1, 256, 0, stream>>>(xa_hf, aWhh_f, abhh_f, nullptr, hT_hf, 0);
  lstm_wmma_kernel<<<1, 256, 0, stream>>>(xa_hb, aWhh_b, abhh_b, nullptr, hT_hb, 1);

  // 8) FC head
  fc1_kernel<<<dim3((2 * H_ + 63) / 64, B_), 64, 0, stream>>>(hT_pf, hT_pb, hT_hf, hT_hb,
                                                              fc1_W, fc1_b, fc1o);
  fc2_kernel<<<B_, 32, 0, stream>>>(fc1o, fc2_W, fc2_b, (float*)d_out);
}